// ScatterHorizontal_40656160424524
// MI455X (gfx1250) — compile-verified
//
#include <hip/hip_runtime.h>
#include <stdint.h>

typedef __attribute__((ext_vector_type(2))) float v2f;
typedef __attribute__((ext_vector_type(8))) float v8f;

namespace {
constexpr int kB    = 1024;
constexpr int kCin  = 64;
constexpr int kCout = 64;
constexpr int kH    = 9;
constexpr int kW    = 9;
constexpr int kHW   = kH * kW;          // 81
constexpr int kNoff = 9;
constexpr int kNpx  = kB * kHW;         // 82944 pixels
constexpr int kPxWG = 32;               // pixels per workgroup (2 N-tiles of 16)
constexpr int kXStr = 68;               // padded LDS channel stride (bank-conflict-free)
}

struct XPtrs { const float* p[kNoff]; };

// Wait for all outstanding async global->LDS transfers (ASYNCcnt == 0).
__device__ __forceinline__ void wait_async0() {
#if __has_builtin(__builtin_amdgcn_s_wait_asynccnt)
  __builtin_amdgcn_s_wait_asynccnt(0);
#else
  asm volatile("s_wait_asynccnt 0x0" ::: "memory");
#endif
}

// Async copy one dword global -> LDS (no VGPR round-trip, ASYNCcnt-tracked).
__device__ __forceinline__ void async_g2l_b32(unsigned lds_addr, const float* gptr) {
  asm volatile("global_load_async_to_lds_b32 %0, %1, off"
               :: "v"(lds_addr), "v"(gptr) : "memory");
}

// Async copy two dwords global -> LDS.
__device__ __forceinline__ void async_g2l_b64(unsigned lds_addr, const float* gptr) {
  asm volatile("global_load_async_to_lds_b64 %0, %1, off"
               :: "v"(lds_addr), "v"(gptr) : "memory");
}

__global__ __launch_bounds__(256)
void ScatterHorizontal_wmma_kernel(XPtrs xp,
                                   const float* __restrict__ weights,
                                   const float* __restrict__ bias,
                                   float* __restrict__ out) {
  // Double-buffered staging: 2*16KB weights + 2*8.5KB activations + 2.25KB bias
  __shared__ float w_lds[2][kCout * kCin];      // A-fragment-ordered W_i
  __shared__ float x_lds[2][kPxWG * kXStr];     // shifted x_i as [px][c]
  __shared__ float bias_lds[kW * kCout];        // pre-combined bias per (w, cout)

  const int tid  = threadIdx.x;
  const int lane = tid & 31;
  const int wav  = tid >> 5;            // 8 waves
  const int m    = wav & 3;             // Cout tile index (4 x 16 = 64)
  const int n0   = (wav >> 2) << 4;     // pixel sub-tile base (0 or 16)
  const int p0   = blockIdx.x * kPxWG;  // first pixel of this workgroup

  // ---- bias: out[b,co,h,w] receives sum of bias[i,co] over offsets i with
  //      0 <= w+4-i < 9, i.e. i in [w-4, w+4] clamped to [0,8] ----
  for (int e = tid; e < kW * kCout; e += 256) {
    const int w  = e >> 6;
    const int co = e & 63;
    const int ilo = (w - 4 < 0) ? 0 : (w - 4);
    const int ihi = (w + 4 > 8) ? 8 : (w + 4);
    float s = 0.f;
    for (int i = ilo; i <= ihi; ++i) s += bias[i * kCout + co];
    bias_lds[e] = s;
  }

  // Re-order W_i[64x64] into V_WMMA_F32_16X16X4_F32 A-fragment order via the
  // async DMA path: each lane copies a contiguous (j=0,1) global pair to a
  // contiguous LDS pair, so the permutation costs zero VGPR traffic.
  // Fragment f = kk*4 + m, element = lane*2 + j:
  //   co = m*16 + (lane&15), k = kk*4 + (lane>>4)*2 + j
  auto stage_weights = [&](int buf, int i) {
    const float* wi = weights + i * (kCout * kCin);
    for (int e2 = tid * 2; e2 < kCout * kCin; e2 += 512) {   // 8 b64 copies/thread
      const int f  = e2 >> 6;
      const int r  = e2 & 63;          // even -> j = 0 (pair covers j=0,1)
      const int l  = r >> 1;
      const int kk = f >> 2;
      const int mm = f & 3;
      const int co = mm * 16 + (l & 15);
      const int k  = kk * 4 + ((l >> 4) << 1);
      const unsigned lds_addr = (unsigned)(uintptr_t)&w_lds[buf][e2];
      async_g2l_b64(lds_addr, wi + co * kCin + k);
    }
  };

  // Gather x_i[b, c, h, w+4-i] into x_lds[px][c] via async global->LDS; lanes
  // whose shifted column falls off the board write a zero through LDS instead
  // (async transfers only EXEC-active lanes). Stride-68 layout keeps the
  // B-fragment ds_load_b64 reads bank-conflict-free.
  auto stage_x = [&](int buf, int i) {
    const float* xi = xp.p[i];
    for (int e = tid; e < kCin * kPxWG; e += 256) {          // 8 copies/thread
      const int c   = e >> 5;
      const int px  = e & 31;
      const int gp  = p0 + px;
      const int b   = gp / kHW;
      const int rem = gp - b * kHW;
      const int h   = rem / kW;
      const int w   = rem - h * kW;
      const int win = w + 4 - i;
      float* dst = &x_lds[buf][px * kXStr + c];
      if (win >= 0 && win < kW) {
        async_g2l_b32((unsigned)(uintptr_t)dst,
                      xi + ((b * kCin + c) * kH + h) * kW + win);
      } else {
        *dst = 0.f;                                          // ds_store, DScnt
      }
    }
  };

  stage_weights(0, 0);
  stage_x(0, 0);
  wait_async0();          // async LDS writes visible before first compute
  __syncthreads();

  v8f acc = {};                             // 16x16 f32 D/C accumulator (8 VGPRs)
  const int c_hi = (lane & 16) ? 2 : 0;     // B fragment: half-wave K sub-pair
  const int pxl  = n0 + (lane & 15);        // this lane's pixel (N) column

  for (int i = 0; i < kNoff; ++i) {
    const int cur = i & 1;
    if (i + 1 < kNoff) {                    // async-prefetch next offset
      stage_weights(cur ^ 1, i + 1);
      stage_x(cur ^ 1, i + 1);
    }
    const float* wb = &w_lds[cur][0];
    const float* xb = &x_lds[cur][0];
#pragma unroll
    for (int kk = 0; kk < 16; ++kk) {       // K = Cin = 64 in steps of 4
      v2f a  = *(const v2f*)(wb + ((kk * 4 + m) << 6) + (lane << 1));
      v2f bm = *(const v2f*)(xb + pxl * kXStr + kk * 4 + c_hi);
      acc = __builtin_amdgcn_wmma_f32_16x16x4_f32(
          false, a, false, bm, (short)0, acc, false, false);
    }
    wait_async0();        // next buffer fully landed in LDS (overlapped w/ wmma)
    __syncthreads();      // all waves done reading 'cur' before it is reused
  }

  // ---- store D + bias; D layout: VGPR r -> M=r (lanes 0-15) / M=r+8 (16-31) ----
  const int gp  = p0 + pxl;
  const int b   = gp / kHW;
  const int rem = gp - b * kHW;
  const int h   = rem / kW;
  const int w   = rem - h * kW;
  const int co0 = m * 16 + ((lane & 16) ? 8 : 0);
  float* op        = out + ((b * kCout + co0) * kH + h) * kW + w;
  const float* bp  = &bias_lds[w * kCout + co0];
#pragma unroll
  for (int r = 0; r < 8; ++r)
    op[r * kHW] = acc[r] + bp[r];
}

extern "C" void kernel_launch(void* const* d_in, const int* in_sizes, int n_in,
                              void* d_out, int out_size, void* d_ws, size_t ws_size,
                              hipStream_t stream) {
  (void)in_sizes; (void)n_in; (void)out_size; (void)d_ws; (void)ws_size;
  XPtrs xp;
  for (int i = 0; i < kNoff; ++i) xp.p[i] = (const float*)d_in[i];
  const float* weights = (const float*)d_in[9];
  const float* bias    = (const float*)d_in[10];
  float* out = (float*)d_out;
  dim3 grid(kNpx / kPxWG);   // 82944 / 32 = 2592 workgroups, no tail
  dim3 block(256);           // 8 wave32 waves
  hipLaunchKernelGGL(ScatterHorizontal_wmma_kernel, grid, block, 0, stream,
                     xp, weights, bias, out);
}